// DBSNLoss_Pretrain_87454124081567
// MI455X (gfx1250) — compile-verified
//
#include <hip/hip_runtime.h>

// ---------------------------------------------------------------------------
// DBSN pretrain loss, MI455X (gfx1250).
// Bandwidth-bound streaming reduction: sigma streams staged into LDS via the
// Tensor Data Mover (double-buffered, TENSORcnt-waited), per-pixel 3x3
// adjugate math in VALU, deterministic two-stage reduction.
// ---------------------------------------------------------------------------

#define PIXELS      (8u * 512u * 512u)     // 2,097,152
#define MNPIX       (1u << 18)             // 512*512 per image
#define TILE_PX     256                    // pixels per tile (1 per thread)
#define TPB         8                      // tiles per block
#define NBLOCKS     (PIXELS / (TILE_PX * TPB))   // 1024
#define NTHREADS    256                    // 8 wave32s
#define MAT_DW      (TILE_PX * 9)          // 2304 dwords per matrix tile
#define TILE_BYTES  (TILE_PX * 36)         // 9216 B per matrix tile
#define TD_ELEMS    (TILE_BYTES / 8)       // 1152 8-byte TDM elements
#define EPSF        1e-6f

typedef unsigned int u32x4 __attribute__((ext_vector_type(4)));
typedef int          i32x4 __attribute__((ext_vector_type(4)));
typedef int          i32x8 __attribute__((ext_vector_type(8)));

// Issue one TDM DMA: copy a contiguous TILE_BYTES chunk from global to LDS.
// 1-D descriptor, data_size = 8 bytes, tile_dim0 = tensor_dim0 = 1152.
__device__ __forceinline__ void tdm_tile_load(const float* gsrc, float* ldst) {
    unsigned long long ga = (unsigned long long)gsrc;
    unsigned lds_off = (unsigned)(unsigned long long)(void*)ldst; // LDS byte offset

    u32x4 g0;
    g0.x = 1u;                                             // count = 1 (valid D#)
    g0.y = lds_off;                                        // lds_addr [63:32]
    g0.z = (unsigned)(ga & 0xFFFFFFFFull);                 // global_addr[31:0]
    g0.w = (unsigned)((ga >> 32) & 0x1FFFFFFull)           // global_addr[56:32]
         | (2u << 30);                                     // type = 2 ("image")

    i32x8 g1;
    g1.s0 = (int)(3u << 16);                               // data_size = 3 (8B), mask=0
    g1.s1 = (int)(((unsigned)TD_ELEMS & 0xFFFFu) << 16);   // tensor_dim0[15:0] @ bits 63:48
    g1.s2 = (int)(((unsigned)TD_ELEMS >> 16) | (1u << 16));// tensor_dim0[31:16] | tensor_dim1=1
    g1.s3 = (int)((unsigned)TD_ELEMS << 16);               // tile_dim0 @ bits 127:112
    g1.s4 = 1;                                             // tile_dim1 = 1, tile_dim2 = 0
    g1.s5 = (int)TD_ELEMS;                                 // tensor_dim0_stride[31:0]
    g1.s6 = (int)(((unsigned)TD_ELEMS & 0xFFFFu) << 16);   // stride0 hi=0 | stride1[15:0]
    g1.s7 = 0;                                             // stride1[47:16]

    i32x4 z4 = (i32x4)0;
#if __clang_major__ >= 23
    i32x8 z8 = (i32x8)0;
    __builtin_amdgcn_tensor_load_to_lds(g0, g1, z4, z4, z8, 0);
#else
    __builtin_amdgcn_tensor_load_to_lds(g0, g1, z4, z4, 0);
#endif
}

// adjugate + determinant of a row-major 3x3
__device__ __forceinline__ void adj3(const float a[9], float A[9], float* det) {
    A[0] = a[4]*a[8] - a[5]*a[7];
    A[1] = a[2]*a[7] - a[1]*a[8];
    A[2] = a[1]*a[5] - a[2]*a[4];
    A[3] = a[5]*a[6] - a[3]*a[8];
    A[4] = a[0]*a[8] - a[2]*a[6];
    A[5] = a[2]*a[3] - a[0]*a[5];
    A[6] = a[3]*a[7] - a[4]*a[6];
    A[7] = a[1]*a[6] - a[0]*a[7];
    A[8] = a[0]*a[4] - a[1]*a[3];
    *det = a[0]*A[0] + a[1]*A[3] + a[2]*A[6];
}

__global__ __launch_bounds__(NTHREADS) void dbsn_partial_kernel(
    const float* __restrict__ target, const float* __restrict__ mu,
    const float* __restrict__ sig_mu, const float* __restrict__ sig_n,
    const float* __restrict__ sig_y,
    float* __restrict__ psum, float* __restrict__ pmax)
{
    __shared__ __align__(16) float sY[2][MAT_DW];
    __shared__ __align__(16) float sN[2][MAT_DW];
    __shared__ __align__(16) float sM[2][MAT_DW];
    __shared__ float rs[NTHREADS];
    __shared__ float rm[NTHREADS];

    const int tid = threadIdx.x;
    const bool wave0 = ((tid >> 5) == 0);       // wave-uniform predicate
    const unsigned tile0 = blockIdx.x * TPB;

    // Prologue: DMA tile 0 into buffer 0 (wave 0 issues; TDM ignores EXEC,
    // but the scalar branch keeps waves 1..7 from fetching these ops).
    if (wave0) {
        unsigned long long off = (unsigned long long)tile0 * TILE_BYTES;
        tdm_tile_load((const float*)((const char*)sig_y  + off), &sY[0][0]);
        tdm_tile_load((const float*)((const char*)sig_n  + off), &sN[0][0]);
        tdm_tile_load((const float*)((const char*)sig_mu + off), &sM[0][0]);
    }

    float acc  = 0.0f;
    float tmax = -1e30f;

    for (int t = 0; t < TPB; ++t) {
        const int buf = t & 1;

        if (wave0) {
            if (t + 1 < TPB) {   // prefetch next tile into the other buffer
                unsigned long long off =
                    (unsigned long long)(tile0 + (unsigned)(t + 1)) * TILE_BYTES;
                const int nb = buf ^ 1;
                tdm_tile_load((const float*)((const char*)sig_y  + off), &sY[nb][0]);
                tdm_tile_load((const float*)((const char*)sig_n  + off), &sN[nb][0]);
                tdm_tile_load((const float*)((const char*)sig_mu + off), &sM[nb][0]);
                // 6 outstanding; tile t's 3 complete first (in-order) when <= 3.
                __builtin_amdgcn_s_wait_tensorcnt(3);
            } else {
                __builtin_amdgcn_s_wait_tensorcnt(0);
            }
        }
        __syncthreads();   // tile t visible to every wave

        // ---- per-pixel math (one pixel per thread) ----
        const unsigned g    = (tile0 + (unsigned)t) * TILE_PX + (unsigned)tid;
        const unsigned bimg = g >> 18;
        const unsigned rem  = g & (MNPIX - 1u);
        const unsigned tb   = bimg * (3u * MNPIX) + rem;

        const float d0 = target[tb]             - mu[tb];
        const float d1 = target[tb + MNPIX]     - mu[tb + MNPIX];
        const float d2 = target[tb + 2u*MNPIX]  - mu[tb + 2u*MNPIX];

        float y[9], n[9], m_[9];
        {
            const float* Yp = &sY[buf][tid * 9];
            const float* Np = &sN[buf][tid * 9];
            const float* Mp = &sM[buf][tid * 9];
#pragma unroll
            for (int k = 0; k < 9; ++k) { y[k] = Yp[k]; n[k] = Np[k]; m_[k] = Mp[k]; }
        }

        // t1 = 0.5 * d^T inv(Sigma_y) d   (via adjugate / det)
        float Ay[9], dety;
        adj3(y, Ay, &dety);
        const float q = d0 * (Ay[0]*d0 + Ay[1]*d1 + Ay[2]*d2)
                      + d1 * (Ay[3]*d0 + Ay[4]*d1 + Ay[5]*d2)
                      + d2 * (Ay[6]*d0 + Ay[7]*d1 + Ay[8]*d2);
        const float t1 = 0.5f * q / dety;

        // t2 = 0.5 * log(max(det(Sigma_n), eps))
        const float detn = n[0]*(n[4]*n[8] - n[5]*n[7])
                         - n[1]*(n[3]*n[8] - n[5]*n[6])
                         + n[2]*(n[3]*n[7] - n[4]*n[6]);
        const float t2 = 0.5f * logf(fmaxf(detn, EPSF));

        // t3 = 0.5 * trace(inv(Sigma_n + eps I) @ Sigma_mu)
        float ne[9];
#pragma unroll
        for (int k = 0; k < 9; ++k) ne[k] = n[k];
        ne[0] += EPSF; ne[4] += EPSF; ne[8] += EPSF;
        float An[9], detne;
        adj3(ne, An, &detne);
        const float tr = An[0]*m_[0] + An[1]*m_[3] + An[2]*m_[6]
                       + An[3]*m_[1] + An[4]*m_[4] + An[5]*m_[7]
                       + An[6]*m_[2] + An[7]*m_[5] + An[8]*m_[8];
        const float t3 = 0.5f * tr / detne;

        acc  += t1 + t2 + t3;
        tmax  = fmaxf(tmax, t1);

        __syncthreads();   // everyone done reading buf before it is refilled
    }

    // ---- deterministic block reduction ----
    rs[tid] = acc;
    rm[tid] = tmax;
    __syncthreads();
#pragma unroll
    for (int s = NTHREADS / 2; s > 0; s >>= 1) {
        if (tid < s) {
            rs[tid] += rs[tid + s];
            rm[tid]  = fmaxf(rm[tid], rm[tid + s]);
        }
        __syncthreads();
    }
    if (tid == 0) {
        psum[blockIdx.x] = rs[0];
        pmax[blockIdx.x] = rm[0];
    }
}

__global__ __launch_bounds__(NTHREADS) void dbsn_final_kernel(
    const float* __restrict__ psum, const float* __restrict__ pmax,
    float* __restrict__ out)
{
    __shared__ float rs[NTHREADS];
    __shared__ float rm[NTHREADS];
    const int tid = threadIdx.x;

    float s = 0.0f, m = -1e30f;
    for (int i = tid; i < NBLOCKS; i += NTHREADS) {
        s += psum[i];
        m  = fmaxf(m, pmax[i]);
    }
    rs[tid] = s;
    rm[tid] = m;
    __syncthreads();
#pragma unroll
    for (int st = NTHREADS / 2; st > 0; st >>= 1) {
        if (tid < st) {
            rs[tid] += rs[tid + st];
            rm[tid]  = fmaxf(rm[tid], rm[tid + st]);
        }
        __syncthreads();
    }
    if (tid == 0) {
        const float loss = rs[0] * (1.0f / (float)PIXELS);
        out[0] = (rm[0] > 1e7f) ? 0.0f : loss;
    }
}

extern "C" void kernel_launch(void* const* d_in, const int* in_sizes, int n_in,
                              void* d_out, int out_size, void* d_ws, size_t ws_size,
                              hipStream_t stream) {
    (void)in_sizes; (void)n_in; (void)out_size; (void)ws_size;
    const float* target = (const float*)d_in[0];
    const float* mu     = (const float*)d_in[1];
    const float* sig_mu = (const float*)d_in[2];
    const float* sig_n  = (const float*)d_in[3];
    const float* sig_y  = (const float*)d_in[4];

    float* psum = (float*)d_ws;
    float* pmax = psum + NBLOCKS;

    dbsn_partial_kernel<<<NBLOCKS, NTHREADS, 0, stream>>>(
        target, mu, sig_mu, sig_n, sig_y, psum, pmax);
    dbsn_final_kernel<<<1, NTHREADS, 0, stream>>>(psum, pmax, (float*)d_out);
}